// MedianFilter3d_82265803588188
// MI455X (gfx1250) — compile-verified
//
#include <hip/hip_runtime.h>
#include <hip/hip_bf16.h>

// ---------------------------------------------------------------------------
// 5x5x5 median filter, 96^3 fp32, replicate pad 2.
// Compute-bound: forgetful selection with pairwise min/max passes and a
// nested-forgetful tail (63->33->17->9->5->3). Buffer of 64 = ceil(125/2)+1
// lives entirely in VGPRs. Data staged global->LDS with CDNA5 async-to-LDS
// copies (ASYNCcnt path, s_wait_asynccnt).
// ---------------------------------------------------------------------------

#define DIM   96
#define TD    4
#define TH    8
#define TW    8
#define PD    (TD + 4)          // 8
#define PH    (TH + 4)          // 12
#define PW    (TW + 4)          // 12
#define TILE  (PD * PH * PW)    // 1152 floats = 4.6 KB LDS
#define NTHREADS 256

#if defined(__has_builtin)
#if __has_builtin(__builtin_amdgcn_global_load_async_to_lds_b32) && \
    __has_builtin(__builtin_amdgcn_s_wait_asynccnt)
#define USE_ASYNC_COPY 1
#endif
#endif

#if defined(USE_ASYNC_COPY)
typedef __attribute__((address_space(1))) int gas_int;   // global
typedef __attribute__((address_space(3))) int las_int;   // LDS
#endif

// Compare-exchange: min -> a, max -> b. Multiset-preserving, bit-exact.
__device__ __forceinline__ void ce(float& a, float& b) {
  float lo = __builtin_fminf(a, b);
  b = __builtin_fmaxf(a, b);
  a = lo;
}

// Pairwise min/max pass over slots [LO..HI]: min lands in A[LO], max in
// A[HI], multiset preserved. Cost ~1.5*S compare-exchanges:
//   1) CE disjoint pairs (lows -> even slots, highs -> odd slots)
//   2) min-chain over lows (+ leftover for odd S) into A[LO]
//   3) max-chain over highs into A[HI]
template <int LO, int HI>
__device__ __forceinline__ void mmpass(float (&A)[64]) {
  constexpr int S = HI - LO + 1;
  constexpr int P = S / 2;
#pragma unroll
  for (int t = 0; t < P; ++t) ce(A[LO + 2 * t], A[LO + 2 * t + 1]);
#pragma unroll
  for (int t = 1; t < P; ++t) ce(A[LO], A[LO + 2 * t]);
  if constexpr ((S & 1) != 0) ce(A[LO], A[HI]);   // leftover joins min side
  constexpr int MT = (S & 1) ? P : (P - 1);
#pragma unroll
  for (int t = 0; t < MT; ++t) ce(A[LO + 2 * t + 1], A[HI]);
}

// Nested forgetful selection: S live elements in slots [0..S), target is the
// middle (rank (S-1)/2). Buffer B = ceil(S/2)+1 occupies slots [0..B); each
// round discards the buffer min (rank < middle) and max (rank > middle) and
// refills from slots B.. ; parity leftover handled by one extra pass.
template <int S>
struct Select {
  static __device__ __forceinline__ float run(float (&A)[64]) {
    constexpr int B = S / 2 + 2;        // == ceil(S/2)+1 for odd S
    constexpr int R = S - B;            // elements beyond the buffer
#pragma unroll
    for (int t = 0; t < R / 2; ++t) {
      mmpass<0, B - 1>(A);
      A[0]     = A[B + 2 * t];
      A[B - 1] = A[B + 2 * t + 1];
    }
    if constexpr ((R & 1) != 0) {
      mmpass<0, B - 1>(A);
      A[0] = A[S - 1];
      return Select<B - 1>::run(A);     // live: slots [0..B-1), size B-1 (odd)
    } else {
      return Select<B>::run(A);         // live: slots [0..B), size B (odd)
    }
  }
};
template <>
struct Select<3> {
  static __device__ __forceinline__ float run(float (&A)[64]) {
    mmpass<0, 2>(A);
    return A[1];
  }
};

// Window element idx in [0,125) -> LDS offset (relative to per-thread base).
// idx = i*25 + j*5 + k over (d,h,w) offsets; order irrelevant for a median.
__device__ __forceinline__ float fetchw(const float* t, int base, int idx) {
  int i = (idx * 41) >> 10;       // idx / 25   (exact for idx < 1000)
  int r = idx - 25 * i;
  int j = (r * 205) >> 10;        // r / 5      (exact for r < 1024)
  int k = r - 5 * j;
  return t[base + (i * PH + j) * PW + k];
}

__device__ __forceinline__ int clamp96(int v) {
  v = v < 0 ? 0 : v;
  return v > (DIM - 1) ? (DIM - 1) : v;
}

__global__ __launch_bounds__(NTHREADS)
void median5x5x5_kernel(const float* __restrict__ in, float* __restrict__ out) {
  __shared__ float tile[TILE];

  const int bw  = blockIdx.x * TW;
  const int bh  = blockIdx.y * TH;
  const int bd  = blockIdx.z * TD;
  const int tid = threadIdx.x;

  // ---- Stage padded 8x12x12 tile into LDS (replicate pad via clamped
  // ---- per-lane addresses). Async global->LDS: per-lane gather into LDS.
  for (int c = tid; c < TILE; c += NTHREADS) {
    int pw = c % PW;
    int t2 = c / PW;
    int ph = t2 % PH;
    int pd = t2 / PH;
    int gw = clamp96(bw + pw - 2);
    int gh = clamp96(bh + ph - 2);
    int gd = clamp96(bd + pd - 2);
    const float* gp = in + ((size_t)gd * DIM + gh) * DIM + gw;
#if defined(USE_ASYNC_COPY)
    __builtin_amdgcn_global_load_async_to_lds_b32(
        (gas_int*)gp, (las_int*)&tile[c], /*offset=*/0, /*cpol=*/0);
#else
    tile[c] = *gp;
#endif
  }
#if defined(USE_ASYNC_COPY)
  __builtin_amdgcn_s_wait_asynccnt(0);   // all of this wave's async copies done
#endif
  __syncthreads();                       // LDS tile visible to whole block

  // ---- Per-thread voxel within the tile.
  const int tw = tid & 7;
  const int th = (tid >> 3) & 7;
  const int td = tid >> 6;
  const int base = (td * PH + th) * PW + tw;

  // ---- Main forgetful phase over the 125-tap window: buffer of 64.
  float A[64];
#pragma unroll
  for (int idx = 0; idx < 64; ++idx) {   // constant-folded offsets
    const int i = idx / 25, r = idx % 25, j = r / 5, k = r % 5;
    A[idx] = tile[base + (i * PH + j) * PW + k];
  }

  // 30 rounds: drop (min,max) -> provably not the median of 125 -> refill 2.
  for (int t2 = 0; t2 < 30; ++t2) {
    mmpass<0, 63>(A);
    const int idx0 = 64 + 2 * t2;
    A[0]  = fetchw(tile, base, idx0);
    A[63] = fetchw(tile, base, idx0 + 1);
  }
  // Consume last element (idx 124 = corner i=j=k=4): 63 live in slots 0..62,
  // having discarded 31 provable-lows and 31 provable-highs -> global median
  // is the middle (rank 31) of these 63.
  mmpass<0, 63>(A);
  A[0] = tile[base + (4 * PH + 4) * PW + 4];

  const float med = Select<63>::run(A);  // 63->33->17->9->5->3

  out[((size_t)(bd + td) * DIM + (bh + th)) * DIM + (bw + tw)] = med;
}

extern "C" void kernel_launch(void* const* d_in, const int* in_sizes, int n_in,
                              void* d_out, int out_size, void* d_ws, size_t ws_size,
                              hipStream_t stream) {
  (void)in_sizes; (void)n_in; (void)out_size; (void)d_ws; (void)ws_size;
  const float* in = (const float*)d_in[0];
  float* out      = (float*)d_out;
  dim3 grid(DIM / TW, DIM / TH, DIM / TD);   // 12 x 12 x 24 = 3456 blocks
  median5x5x5_kernel<<<grid, dim3(NTHREADS), 0, stream>>>(in, out);
}